// GraphAttentionV3_10642928959906
// MI455X (gfx1250) — compile-verified
//
#include <hip/hip_runtime.h>
#include <math.h>

// ---------------- problem constants (from reference) ----------------
#define BGR   128      // graphs
#define NPER  64       // nodes per graph
#define EUND  256      // kept/sorted tokens per graph
#define NTOT  8192     // total nodes
#define EDIR  65536    // directed edges
#define EKEEP 32768    // kept edges (tokens)
#define FIN   5
#define H1D   256
#define H2D   512
#define ATTD  1536
#define EPSV  1e-5f

typedef __attribute__((ext_vector_type(2))) float v2f;
typedef __attribute__((ext_vector_type(8))) float v8f;

__device__ inline v8f wmma4f32(v2f a, v2f b, v8f c) {
  // D = A(16x4 f32) x B(4x16 f32) + C(16x16 f32) -> v_wmma_f32_16x16x4_f32
  return __builtin_amdgcn_wmma_f32_16x16x4_f32(false, a, false, b, (short)0, c,
                                               false, false);
}

// ---------------- small utility kernels ----------------
__global__ void zero_kernel(float* __restrict__ p, size_t n) {
  size_t i = (size_t)blockIdx.x * blockDim.x + threadIdx.x;
  if (i < n) p[i] = 0.f;
}

__global__ void edge_w_kernel(const float* __restrict__ ea,
                              const float* __restrict__ wW,
                              const float* __restrict__ wb,
                              float* __restrict__ w) {
  int e = blockIdx.x * blockDim.x + threadIdx.x;
  if (e >= EDIR) return;
  float a0 = ea[2 * e], a1 = ea[2 * e + 1];
  float v = a0 * wW[0] + ((a1 < 0.5f) ? wW[1] : wW[2]) + wb[0];
  w[e] = fmaxf(v, 0.f);
}

__global__ void agg1_kernel(const int* __restrict__ edges,
                            const float* __restrict__ w,
                            const float* __restrict__ x,
                            float* __restrict__ agg1) {
  int e = blockIdx.x * blockDim.x + threadIdx.x;
  if (e >= EDIR) return;
  int s = edges[e], d = edges[EDIR + e];
  float we = w[e];
#pragma unroll
  for (int f = 0; f < FIN; ++f)
    atomicAdd(&agg1[d * FIN + f], we * x[s * FIN + f]);
}

__global__ void gconv1_kernel(const float* __restrict__ agg1,
                              const float* __restrict__ x,
                              const float* __restrict__ Wrel,
                              const float* __restrict__ brel,
                              const float* __restrict__ Wroot,
                              float* __restrict__ h1) {
  int n = blockIdx.x, j = threadIdx.x;
  float acc = brel[j];
#pragma unroll
  for (int f = 0; f < FIN; ++f) {
    acc += agg1[n * FIN + f] * Wrel[f * H1D + j];
    acc += x[n * FIN + f] * Wroot[f * H1D + j];
  }
  h1[(size_t)n * H1D + j] = fmaxf(acc, 0.f);
}

__global__ void agg2_kernel(const int* __restrict__ edges,
                            const float* __restrict__ w,
                            const float* __restrict__ h1,
                            float* __restrict__ agg2) {
  int e = blockIdx.x;
  int c = threadIdx.x;
  int s = edges[e], d = edges[EDIR + e];
  float we = w[e];
  atomicAdd(&agg2[(size_t)d * H1D + c], we * h1[(size_t)s * H1D + c]);
}

__global__ void graphnorm_kernel(float* __restrict__ h,
                                 const float* __restrict__ gw,
                                 const float* __restrict__ gb,
                                 const float* __restrict__ gms) {
  int b = blockIdx.x;
  int base = b * NPER;
  for (int c = threadIdx.x; c < H2D; c += blockDim.x) {
    float s = 0.f;
    for (int n = 0; n < NPER; ++n) s += h[(size_t)(base + n) * H2D + c];
    float m = gms[c] * (s / (float)NPER);
    float v = 0.f;
    for (int n = 0; n < NPER; ++n) {
      float ce = h[(size_t)(base + n) * H2D + c] - m;
      v += ce * ce;
    }
    float inv = rsqrtf(v / (float)NPER + EPSV);
    for (int n = 0; n < NPER; ++n) {
      float ce = h[(size_t)(base + n) * H2D + c] - m;
      h[(size_t)(base + n) * H2D + c] = gw[c] * ce * inv + gb[c];
    }
  }
}

// per-graph stable sort of kept edges (second half of each graph's edge list,
// where src=hi > dst=lo). key = (src*NTOT+dst)<<8 | j (stability in low bits)
__global__ void sort_edges_kernel(const int* __restrict__ edges,
                                  const float* __restrict__ ea,
                                  int* __restrict__ esrc, int* __restrict__ edst,
                                  float* __restrict__ ea0, float* __restrict__ ecl) {
  __shared__ unsigned long long key[EUND];
  int b = blockIdx.x, t = threadIdx.x;
  int e = b * (2 * EUND) + EUND + t;
  unsigned s = (unsigned)edges[e];
  unsigned d = (unsigned)edges[EDIR + e];
  key[t] = (((unsigned long long)(s * (unsigned)NTOT + d)) << 8) |
           (unsigned long long)t;
  __syncthreads();
  for (int k = 2; k <= EUND; k <<= 1) {
    for (int j = k >> 1; j > 0; j >>= 1) {
      int p = t ^ j;
      if (p > t) {
        bool up = ((t & k) == 0);
        unsigned long long a = key[t], bb = key[p];
        bool sw = up ? (a > bb) : (a < bb);
        if (sw) { key[t] = bb; key[p] = a; }
      }
      __syncthreads();
    }
  }
  int j0 = (int)(key[t] & 255ull);
  int e2 = b * (2 * EUND) + EUND + j0;
  int o = b * EUND + t;
  esrc[o] = edges[e2];
  edst[o] = edges[EDIR + e2];
  ea0[o] = ea[2 * e2];
  ecl[o] = ea[2 * e2 + 1];
}

// feat[r] = [h2[src], relu(edge_emb), h2[dst]]
__global__ void feat_kernel(const float* __restrict__ h2,
                            const int* __restrict__ esrc,
                            const int* __restrict__ edst,
                            const float* __restrict__ ea0,
                            const float* __restrict__ ecl,
                            const float* __restrict__ eW,
                            const float* __restrict__ eb,
                            float* __restrict__ feat) {
  int r = blockIdx.x;
  int s = esrc[r], d = edst[r];
  float a0 = ea0[r], cls = ecl[r];
  for (int c = threadIdx.x; c < ATTD; c += blockDim.x) {
    float v;
    if (c < H2D) {
      v = h2[(size_t)s * H2D + c];
    } else if (c < 2 * H2D) {
      int cc = c - H2D;
      float e = a0 * eW[cc] + ((cls < 0.5f) ? eW[H2D + cc] : eW[2 * H2D + cc]) +
                eb[cc];
      v = fmaxf(e, 0.f);
    } else {
      v = h2[(size_t)d * H2D + (c - 2 * H2D)];
    }
    feat[(size_t)r * ATTD + c] = v;
  }
}

__global__ void fuse_bias_kernel(const float* __restrict__ vec,
                                 const float* __restrict__ W,
                                 const float* __restrict__ b,
                                 float* __restrict__ out) {
  int j = blockIdx.x * blockDim.x + threadIdx.x;
  if (j >= ATTD) return;
  float acc = 0.f;
  for (int i = 0; i < ATTD; ++i) acc += vec[i] * W[(size_t)i * ATTD + j];
  out[j] = acc + b[j];
}

__global__ void matvec_row_kernel(const float* __restrict__ W,
                                  const float* __restrict__ vec,
                                  float* __restrict__ out) {
  int d = blockIdx.x * blockDim.x + threadIdx.x;
  if (d >= ATTD) return;
  float acc = 0.f;
  for (int j = 0; j < ATTD; ++j) acc += W[(size_t)d * ATTD + j] * vec[j];
  out[d] = acc;
}

__global__ void dot_kernel(const float* __restrict__ a,
                           const float* __restrict__ b,
                           float* __restrict__ out) {
  __shared__ float red[256];
  int t = threadIdx.x;
  float s = 0.f;
  for (int i = t; i < ATTD; i += 256) s += a[i] * b[i];
  red[t] = s;
  __syncthreads();
  for (int o = 128; o > 0; o >>= 1) {
    if (t < o) red[t] += red[t + o];
    __syncthreads();
  }
  if (t == 0) out[0] = red[0];
}

__global__ __launch_bounds__(256) void fufg_kernel(const float* __restrict__ feat,
                                                   const float* __restrict__ u,
                                                   const float* __restrict__ g,
                                                   float* __restrict__ fu,
                                                   float* __restrict__ fg) {
  __shared__ float red[256];
  int r = blockIdx.x, t = threadIdx.x;
  float su = 0.f, sg = 0.f;
  for (int c = t; c < ATTD; c += 256) {
    float f = feat[(size_t)r * ATTD + c];
    su += f * u[c];
    sg += f * g[c];
  }
  red[t] = su; __syncthreads();
  for (int o = 128; o > 0; o >>= 1) { if (t < o) red[t] += red[t + o]; __syncthreads(); }
  if (t == 0) fu[r] = red[0];
  __syncthreads();
  red[t] = sg; __syncthreads();
  for (int o = 128; o > 0; o >>= 1) { if (t < o) red[t] += red[t + o]; __syncthreads(); }
  if (t == 0) fg[r] = red[0];
}

// ---------------- generic fp32 WMMA GEMM (compile-time variants) ----------------
// C[M,N] = op(A[M,K] @ B) (+bias[N]) (+add[M,N]) (relu?)
// grid = (N/64, M/128), 256 threads (8 waves), each wave computes 16x64.
// All flags are template params -> branch-free inner loop of b64 loads + wmma.
template <bool TRANSB, bool BIAS, bool ADD, bool RELU>
__global__ __launch_bounds__(256) void gemm_wmma_f32(
    const float* __restrict__ A, int lda,
    const float* __restrict__ Bm, int ldb,
    const float* __restrict__ bias,
    const float* __restrict__ add, int ldadd,
    float* __restrict__ C, int ldc, int K) {
  const int lane = threadIdx.x & 31;
  const int wave = threadIdx.x >> 5;
  const int l16 = lane & 15;
  const int lh = lane >> 4;
  const int m0 = blockIdx.y * 128 + wave * 16;
  const int nb = blockIdx.x * 64;
  v8f acc[4];
#pragma unroll
  for (int t = 0; t < 4; ++t)
#pragma unroll
    for (int r = 0; r < 8; ++r) acc[t][r] = 0.f;

  // hoisted base pointers; all pair loads are 8B-aligned (lda/ldb even, k0%4==0)
  const float* Ap = A + (size_t)(m0 + l16) * lda + 2 * lh;
  const float* Bt0;
  const float* Bt1;
  const float* Bt2;
  const float* Bt3;
  const float* Bc;
  if (TRANSB) {
    Bt0 = Bm + (size_t)(nb + 0 * 16 + l16) * ldb + 2 * lh;
    Bt1 = Bm + (size_t)(nb + 1 * 16 + l16) * ldb + 2 * lh;
    Bt2 = Bm + (size_t)(nb + 2 * 16 + l16) * ldb + 2 * lh;
    Bt3 = Bm + (size_t)(nb + 3 * 16 + l16) * ldb + 2 * lh;
    Bc = nullptr;
  } else {
    Bc = Bm + (size_t)(2 * lh) * ldb + nb + l16;
    Bt0 = Bt1 = Bt2 = Bt3 = nullptr;
  }

  for (int k0 = 0; k0 < K; k0 += 4) {
    v2f a = *(const v2f*)(Ap + k0);
    if (TRANSB) {
      v2f b0 = *(const v2f*)(Bt0 + k0);
      v2f b1 = *(const v2f*)(Bt1 + k0);
      v2f b2 = *(const v2f*)(Bt2 + k0);
      v2f b3 = *(const v2f*)(Bt3 + k0);
      acc[0] = wmma4f32(a, b0, acc[0]);
      acc[1] = wmma4f32(a, b1, acc[1]);
      acc[2] = wmma4f32(a, b2, acc[2]);
      acc[3] = wmma4f32(a, b3, acc[3]);
    } else {
      const float* Br = Bc + (size_t)k0 * ldb;
#pragma unroll
      for (int t = 0; t < 4; ++t) {
        v2f b;
        b.x = Br[t * 16];
        b.y = Br[(size_t)ldb + t * 16];
        acc[t] = wmma4f32(a, b, acc[t]);
      }
    }
  }
#pragma unroll
  for (int t = 0; t < 4; ++t) {
    int col = nb + t * 16 + l16;
#pragma unroll
    for (int r = 0; r < 8; ++r) {
      int row = m0 + r + 8 * lh;
      float v = acc[t][r];
      if (BIAS) v += bias[col];
      if (ADD) v += add[(size_t)row * ldadd + col];
      if (RELU) v = fmaxf(v, 0.f);
      C[(size_t)row * ldc + col] = v;
    }
  }
}

static void launch_gemm(dim3 grid, hipStream_t st, const float* A, int lda,
                        const float* B, int ldb, bool transB,
                        const float* bias, const float* add, int ldadd,
                        float* C, int ldc, int K, bool relu) {
  if (transB) {
    gemm_wmma_f32<true, false, false, false><<<grid, 256, 0, st>>>(
        A, lda, B, ldb, nullptr, nullptr, 0, C, ldc, K);
  } else if (bias && add && relu) {
    gemm_wmma_f32<false, true, true, true><<<grid, 256, 0, st>>>(
        A, lda, B, ldb, bias, add, ldadd, C, ldc, K);
  } else if (bias && add) {
    gemm_wmma_f32<false, true, true, false><<<grid, 256, 0, st>>>(
        A, lda, B, ldb, bias, add, ldadd, C, ldc, K);
  } else {
    gemm_wmma_f32<false, false, false, false><<<grid, 256, 0, st>>>(
        A, lda, B, ldb, nullptr, nullptr, 0, C, ldc, K);
  }
}

// ---------------- fused per-graph attention ----------------
// scores = scale*(T_b @ feat_b^T + fu_i + fg_j + c) -> softmax -> PF = P@feat_b
// PF overwrites T_b region. LDS holds the full 256x256 score matrix.
// Row stride 258 keeps ds_load_b64 alignment and is bank-conflict benign.
#define SLD 258
__global__ __launch_bounds__(256) void attention_kernel(
    const float* __restrict__ feat, float* __restrict__ Tbuf,
    const float* __restrict__ fu, const float* __restrict__ fg,
    const float* __restrict__ cbuf,
    const int* __restrict__ esrc, const int* __restrict__ edst) {
  extern __shared__ float S[];
  const int b = blockIdx.x;
  const int gb = b * EUND;
  const int lane = threadIdx.x & 31;
  const int wave = threadIdx.x >> 5;
  const int l16 = lane & 15;
  const int lh = lane >> 4;
  const float scale = rsqrtf((float)ATTD);

  // phase 1: raw dot products T_b @ feat_b^T into LDS
  for (int tile = wave; tile < 256; tile += 8) {
    int ti = tile >> 4, tj = tile & 15;
    v8f acc;
#pragma unroll
    for (int r = 0; r < 8; ++r) acc[r] = 0.f;
    const float* Ap = Tbuf + (size_t)(gb + ti * 16 + l16) * ATTD + 2 * lh;
    const float* Bp = feat + (size_t)(gb + tj * 16 + l16) * ATTD + 2 * lh;
    for (int k0 = 0; k0 < ATTD; k0 += 4) {
      v2f a = *(const v2f*)(Ap + k0);
      v2f bb = *(const v2f*)(Bp + k0);
      acc = wmma4f32(a, bb, acc);
    }
#pragma unroll
    for (int r = 0; r < 8; ++r)
      S[(ti * 16 + r + 8 * lh) * SLD + tj * 16 + l16] = acc[r];
  }
  __syncthreads();

  // phase 2: rank-1 bias terms, scale, mask, softmax (one row per thread)
  {
    int i = threadIdx.x;
    float cc = cbuf[0];
    float fui = fu[gb + i];
    float mx = -__builtin_inff();
    for (int j = 0; j < EUND; ++j) {
      float s = scale * (S[i * SLD + j] + fui + fg[gb + j] + cc);
      if (esrc[gb + j] == edst[gb + j]) s = -__builtin_inff();
      S[i * SLD + j] = s;
      mx = fmaxf(mx, s);
    }
    float sum = 0.f;
    for (int j = 0; j < EUND; ++j) {
      float e = __expf(S[i * SLD + j] - mx);
      S[i * SLD + j] = e;
      sum += e;
    }
    float inv = 1.f / sum;
    for (int j = 0; j < EUND; ++j) S[i * SLD + j] *= inv;
  }
  __syncthreads();

  // phase 3: PF = P @ feat_b  (overwrite T_b in global)
  for (int tile = wave; tile < 16 * 96; tile += 8) {
    int ti = tile / 96, tn = tile % 96;
    int n = tn * 16 + l16;
    v8f acc;
#pragma unroll
    for (int r = 0; r < 8; ++r) acc[r] = 0.f;
    const float* Sp = S + (ti * 16 + l16) * SLD + 2 * lh;  // even offset: b64
    const float* Bp = feat + (size_t)(gb + 2 * lh) * ATTD + n;
    for (int k0 = 0; k0 < EUND; k0 += 4) {
      v2f a = *(const v2f*)(Sp + k0);
      v2f bb;
      const float* Br = Bp + (size_t)k0 * ATTD;
      bb.x = Br[0];
      bb.y = Br[ATTD];
      acc = wmma4f32(a, bb, acc);
    }
#pragma unroll
    for (int r = 0; r < 8; ++r)
      Tbuf[(size_t)(gb + ti * 16 + r + 8 * lh) * ATTD + n] = acc[r];
  }
}

// LayerNorm + head projection, one block per token row
__global__ __launch_bounds__(256) void ln_head_kernel(
    const float* __restrict__ res, const float* __restrict__ lng,
    const float* __restrict__ lnb, const float* __restrict__ hW,
    const float* __restrict__ hb, float* __restrict__ out) {
  __shared__ float red[256];
  int r = blockIdx.x, t = threadIdx.x;
  float x[6];
  float s = 0.f;
#pragma unroll
  for (int q = 0; q < 6; ++q) {
    x[q] = res[(size_t)r * ATTD + t + q * 256];
    s += x[q];
  }
  red[t] = s; __syncthreads();
  for (int o = 128; o > 0; o >>= 1) { if (t < o) red[t] += red[t + o]; __syncthreads(); }
  float mu = red[0] / (float)ATTD;
  __syncthreads();
  float v = 0.f;
#pragma unroll
  for (int q = 0; q < 6; ++q) { float d = x[q] - mu; v += d * d; }
  red[t] = v; __syncthreads();
  for (int o = 128; o > 0; o >>= 1) { if (t < o) red[t] += red[t + o]; __syncthreads(); }
  float inv = rsqrtf(red[0] / (float)ATTD + EPSV);
  __syncthreads();
  float dacc = 0.f;
#pragma unroll
  for (int q = 0; q < 6; ++q) {
    int c = t + q * 256;
    float y = lng[c] * (x[q] - mu) * inv + lnb[c];
    dacc += y * hW[c];
  }
  red[t] = dacc; __syncthreads();
  for (int o = 128; o > 0; o >>= 1) { if (t < o) red[t] += red[t + o]; __syncthreads(); }
  if (t == 0) out[r] = red[0] + hb[0];
}

// pairwise argmin + output assembly
__global__ void final_kernel(const float* __restrict__ hout,
                             const int* __restrict__ esrc,
                             const int* __restrict__ edst,
                             const float* __restrict__ ecl,
                             float* __restrict__ out) {
  int idx = blockIdx.x * blockDim.x + threadIdx.x;
  if (idx >= BGR * (EUND / 2)) return;
  int b = idx / (EUND / 2), i = idx % (EUND / 2);
  int rb = b * EUND + 2 * i;
  float o0 = hout[rb], o1 = hout[rb + 1];
  int mi = (o1 < o0) ? 1 : 0;
  out[(size_t)idx * 2 + 0] = (float)esrc[rb];
  out[(size_t)idx * 2 + 1] = (float)edst[rb];
  out[EKEEP + idx] = mi ? o1 : o0;                      // val
  out[EKEEP + BGR * (EUND / 2) + idx] = ecl[rb + mi];   // cls
}

// ---------------- host launcher ----------------
extern "C" void kernel_launch(void* const* d_in, const int* in_sizes, int n_in,
                              void* d_out, int out_size, void* d_ws,
                              size_t ws_size, hipStream_t stream) {
  const float* x        = (const float*)d_in[0];
  const int*   edges    = (const int*)d_in[1];
  const float* ea       = (const float*)d_in[2];
  const float* w_emb_W  = (const float*)d_in[5];
  const float* w_emb_b  = (const float*)d_in[6];
  const float* gc1_Wrel = (const float*)d_in[7];
  const float* gc1_brel = (const float*)d_in[8];
  const float* gc1_Wroot= (const float*)d_in[9];
  const float* gc2_Wrel = (const float*)d_in[10];
  const float* gc2_brel = (const float*)d_in[11];
  const float* gc2_Wroot= (const float*)d_in[12];
  const float* gn_w     = (const float*)d_in[13];
  const float* gn_b     = (const float*)d_in[14];
  const float* gn_ms    = (const float*)d_in[15];
  const float* eemb_W   = (const float*)d_in[16];
  const float* eemb_b   = (const float*)d_in[17];
  const float* qkv_W    = (const float*)d_in[18];
  const float* qkv_b    = (const float*)d_in[19];
  const float* in_Wq    = (const float*)d_in[20];
  const float* in_Wk    = (const float*)d_in[21];
  const float* in_Wv    = (const float*)d_in[22];
  const float* in_bq    = (const float*)d_in[23];
  const float* in_bk    = (const float*)d_in[24];
  const float* in_bv    = (const float*)d_in[25];
  const float* out_W    = (const float*)d_in[26];
  const float* out_b    = (const float*)d_in[27];
  const float* ln_g     = (const float*)d_in[28];
  const float* ln_b     = (const float*)d_in[29];
  const float* head_W   = (const float*)d_in[30];
  const float* head_b   = (const float*)d_in[31];
  float* out = (float*)d_out;

  // workspace carve (~503 MB f32)
  float* W = (float*)d_ws;
  size_t o = 0;
  auto alloc = [&](size_t n) { float* p = W + o; o += n; return p; };
  float* w_e  = alloc(EDIR);
  float* agg1 = alloc((size_t)NTOT * FIN);
  float* h1   = alloc((size_t)NTOT * H1D);
  float* agg2 = alloc((size_t)NTOT * H1D);
  float* tmp  = alloc((size_t)NTOT * H2D);
  float* h2   = alloc((size_t)NTOT * H2D);
  int*   esrc = (int*)alloc(EKEEP);
  int*   edst = (int*)alloc(EKEEP);
  float* ea0  = alloc(EKEEP);
  float* ecl  = alloc(EKEEP);
  float* Wqf  = alloc((size_t)ATTD * ATTD);
  float* Wkf  = alloc((size_t)ATTD * ATTD);
  float* Wvf  = alloc((size_t)ATTD * ATTD);
  float* Amid = alloc((size_t)ATTD * ATTD);
  float* Wvo  = alloc((size_t)ATTD * ATTD);
  float* bqf  = alloc(ATTD);
  float* bkf  = alloc(ATTD);
  float* bvf  = alloc(ATTD);
  float* uvec = alloc(ATTD);
  float* gvec = alloc(ATTD);
  float* bo   = alloc(ATTD);
  float* cbuf = alloc(1);
  float* feat = alloc((size_t)EKEEP * ATTD);
  float* Tbuf = alloc((size_t)EKEEP * ATTD);
  float* fu   = alloc(EKEEP);
  float* fg   = alloc(EKEEP);
  float* hout = alloc(EKEEP);
  (void)ws_size; (void)n_in; (void)in_sizes; (void)out_size;

  // ---- graph convolutions ----
  {
    size_t n1 = (size_t)NTOT * FIN, n2 = (size_t)NTOT * H1D;
    zero_kernel<<<(unsigned)((n1 + 255) / 256), 256, 0, stream>>>(agg1, n1);
    zero_kernel<<<(unsigned)((n2 + 255) / 256), 256, 0, stream>>>(agg2, n2);
  }
  edge_w_kernel<<<EDIR / 256, 256, 0, stream>>>(ea, w_emb_W, w_emb_b, w_e);
  agg1_kernel<<<EDIR / 256, 256, 0, stream>>>(edges, w_e, x, agg1);
  gconv1_kernel<<<NTOT, H1D, 0, stream>>>(agg1, x, gc1_Wrel, gc1_brel,
                                          gc1_Wroot, h1);
  agg2_kernel<<<EDIR, H1D, 0, stream>>>(edges, w_e, h1, agg2);
  // tmp = h1 @ gc2_Wroot
  launch_gemm(dim3(H2D / 64, NTOT / 128), stream, h1, H1D, gc2_Wroot, H2D,
              false, nullptr, nullptr, 0, tmp, H2D, H1D, false);
  // h2 = relu(agg2 @ gc2_Wrel + brel + tmp)
  launch_gemm(dim3(H2D / 64, NTOT / 128), stream, agg2, H1D, gc2_Wrel, H2D,
              false, gc2_brel, tmp, H2D, h2, H2D, H1D, true);
  graphnorm_kernel<<<BGR, 256, 0, stream>>>(h2, gn_w, gn_b, gn_ms);

  // ---- edge select + stable sort + features ----
  sort_edges_kernel<<<BGR, EUND, 0, stream>>>(edges, ea, esrc, edst, ea0, ecl);
  feat_kernel<<<EKEEP, 256, 0, stream>>>(h2, esrc, edst, ea0, ecl, eemb_W,
                                         eemb_b, feat);

  // ---- fuse qkv/in/out linear layers ----
  launch_gemm(dim3(ATTD / 64, ATTD / 128), stream, qkv_W + 0, 3 * ATTD, in_Wq,
              ATTD, false, nullptr, nullptr, 0, Wqf, ATTD, ATTD, false);
  launch_gemm(dim3(ATTD / 64, ATTD / 128), stream, qkv_W + ATTD, 3 * ATTD,
              in_Wk, ATTD, false, nullptr, nullptr, 0, Wkf, ATTD, ATTD, false);
  launch_gemm(dim3(ATTD / 64, ATTD / 128), stream, qkv_W + 2 * ATTD, 3 * ATTD,
              in_Wv, ATTD, false, nullptr, nullptr, 0, Wvf, ATTD, ATTD, false);
  fuse_bias_kernel<<<ATTD / 256, 256, 0, stream>>>(qkv_b + 0, in_Wq, in_bq, bqf);
  fuse_bias_kernel<<<ATTD / 256, 256, 0, stream>>>(qkv_b + ATTD, in_Wk, in_bk, bkf);
  fuse_bias_kernel<<<ATTD / 256, 256, 0, stream>>>(qkv_b + 2 * ATTD, in_Wv, in_bv, bvf);
  // Amid = Wq_f @ Wk_f^T ; Wvo = Wv_f @ out_W
  launch_gemm(dim3(ATTD / 64, ATTD / 128), stream, Wqf, ATTD, Wkf, ATTD, true,
              nullptr, nullptr, 0, Amid, ATTD, ATTD, false);
  launch_gemm(dim3(ATTD / 64, ATTD / 128), stream, Wvf, ATTD, out_W, ATTD,
              false, nullptr, nullptr, 0, Wvo, ATTD, ATTD, false);
  matvec_row_kernel<<<ATTD / 256, 256, 0, stream>>>(Wqf, bkf, uvec);
  matvec_row_kernel<<<ATTD / 256, 256, 0, stream>>>(Wkf, bqf, gvec);
  dot_kernel<<<1, 256, 0, stream>>>(bqf, bkf, cbuf);
  fuse_bias_kernel<<<ATTD / 256, 256, 0, stream>>>(bvf, out_W, out_b, bo);

  // ---- attention ----
  fufg_kernel<<<EKEEP, 256, 0, stream>>>(feat, uvec, gvec, fu, fg);
  // T = feat @ Amid
  launch_gemm(dim3(ATTD / 64, EKEEP / 128), stream, feat, ATTD, Amid, ATTD,
              false, nullptr, nullptr, 0, Tbuf, ATTD, ATTD, false);
  attention_kernel<<<BGR, 256, (size_t)EUND * SLD * sizeof(float), stream>>>(
      feat, Tbuf, fu, fg, cbuf, esrc, edst);
  // feat = feat + PF @ Wvo + bo   (residual + out-proj)
  launch_gemm(dim3(ATTD / 64, EKEEP / 128), stream, Tbuf, ATTD, Wvo, ATTD,
              false, bo, feat, ATTD, feat, ATTD, ATTD, false);

  // ---- layernorm + head + pairing ----
  ln_head_kernel<<<EKEEP, 256, 0, stream>>>(feat, ln_g, ln_b, head_W, head_b,
                                            hout);
  final_kernel<<<(BGR * (EUND / 2) + 255) / 256, 256, 0, stream>>>(
      hout, esrc, edst, ecl, out);
}